// DQ_CA_21148418965827
// MI455X (gfx1250) — compile-verified
//
#include <hip/hip_runtime.h>
#include <hip/hip_bf16.h>

typedef __attribute__((ext_vector_type(2))) float v2f;
typedef __attribute__((ext_vector_type(8))) float v8f;

#define NHEAD 8
#define DHEAD 32
#define BATCH 4
#define QLEN  200
#define NMEM  16384
#define CDIM  256
#define QPAD  208   // 13 tiles of 16
#define QSLOT 256   // stride for per-(b,h) best slots
#define ROWS_PER_WG 128

__device__ __forceinline__ unsigned enc_f32(float v) {
  unsigned u = __float_as_uint(v);
  return (u & 0x80000000u) ? ~u : (u | 0x80000000u);
}

__device__ __forceinline__ float block_reduce_sum(float v, float* red) {
  int c = threadIdx.x;
  red[c] = v;
  __syncthreads();
  for (int s = 128; s > 0; s >>= 1) {
    if (c < s) red[c] += red[c + s];
    __syncthreads();
  }
  float r = red[0];
  __syncthreads();
  return r;
}

// ---------------- qn = LayerNorm(q); key = l2norm(qn + query_pos) ----------------
__global__ void __launch_bounds__(256)
prep_q_kernel(const float* __restrict__ q, const float* __restrict__ qpos,
              const float* __restrict__ gamma, const float* __restrict__ beta,
              float* __restrict__ qn, float* __restrict__ keyb) {
  __shared__ float red[256];
  size_t t = blockIdx.x;            // token index b*QLEN + q
  int c = threadIdx.x;
  float v = q[t * CDIM + c];
  float mean = block_reduce_sum(v, red) * (1.0f / CDIM);
  float d = v - mean;
  float var = block_reduce_sum(d * d, red) * (1.0f / CDIM);
  float qnv = d / sqrtf(var + 1e-5f) * gamma[c] + beta[c];
  qn[t * CDIM + c] = qnv;
  float k2 = qnv + qpos[t * CDIM + c];
  float ss = block_reduce_sum(k2 * k2, red);
  keyb[t * CDIM + c] = k2 / fmaxf(sqrtf(ss), 1e-12f);
}

// ---------------- xhat = l2norm(x + x_pos) ----------------
__global__ void __launch_bounds__(256)
prep_x_kernel(const float* __restrict__ x, const float* __restrict__ xpos,
              float* __restrict__ xhat) {
  __shared__ float red[256];
  size_t t = blockIdx.x;            // token index b*NMEM + n
  int c = threadIdx.x;
  float v = x[t * CDIM + c] + xpos[t * CDIM + c];
  float ss = block_reduce_sum(v * v, red);
  xhat[t * CDIM + c] = v / fmaxf(sqrtf(ss), 1e-12f);
}

// ---------------- sim GEMM (WMMA f32 16x16x4) + mask + argmax ----------------
__global__ void __launch_bounds__(256)
sim_kernel(const float* __restrict__ xhat, const float* __restrict__ keyb,
           const unsigned char* __restrict__ mask,     // [B*H, Q, N] bool
           float* __restrict__ sim,                    // [B,H,N,Q]
           unsigned long long* __restrict__ best) {    // [B*H, QSLOT] packed
  __shared__ float keyLds[QPAD * DHEAD];
  __shared__ unsigned long long bestLds[QPAD];
  int bh = blockIdx.x;
  int b = bh >> 3, h = bh & 7;
  int chunk = blockIdx.y * ROWS_PER_WG;
  int tid = threadIdx.x;

  // stage key_h slice [QPAD x 32] into LDS (zero pad q>=200)
  for (int i = tid; i < QPAD * DHEAD; i += 256) {
    int qq = i / DHEAD, d = i - qq * DHEAD;
    keyLds[i] = (qq < QLEN)
        ? keyb[((size_t)(b * QLEN + qq)) * CDIM + h * DHEAD + d] : 0.0f;
  }
  if (tid < QPAD) bestLds[tid] = 0ull;
  __syncthreads();

  int wave = tid >> 5;
  int lane = tid & 31;
  int lm = lane & 15;
  int hi = lane >> 4;
  int n0 = chunk + wave * 16;

  // A fragments: A[m,k] = xhat[n0+m, h*32 + k]  (16x4 f32 layout: lane=M, hi half: K+2)
  const float* arow = xhat + ((size_t)b * NMEM + (n0 + lm)) * CDIM + h * DHEAD;
  v2f a[8];
#pragma unroll
  for (int j = 0; j < 8; ++j) {
    a[j].x = arow[4 * j + 2 * hi];
    a[j].y = arow[4 * j + 2 * hi + 1];
  }

  const unsigned char* mBase = mask + (size_t)bh * QLEN * NMEM;
  float* simBase = sim + (size_t)bh * NMEM * QLEN;

  for (int t13 = 0; t13 < 13; ++t13) {
    int q0 = t13 * 16;
    // B fragments: B[k, ncol] = key[q0+ncol, kbase+k] (symmetric layout)
    v2f bf[8];
#pragma unroll
    for (int j = 0; j < 8; ++j) {
      const float* krow = keyLds + (q0 + lm) * DHEAD;
      bf[j].x = krow[4 * j + 2 * hi];
      bf[j].y = krow[4 * j + 2 * hi + 1];
    }
    v8f acc = {};
#pragma unroll
    for (int j = 0; j < 8; ++j)
      acc = __builtin_amdgcn_wmma_f32_16x16x4_f32(
          false, a[j], false, bf[j], (short)0, acc, false, false);

    // C/D layout: lane owns column q = q0+lm, rows m = r + 8*hi
    int qc = q0 + lm;
    if (qc < QLEN) {
      const unsigned char* mrow = mBase + (size_t)qc * NMEM + n0 + 8 * hi;
      unsigned long long lbest = 0ull;
#pragma unroll
      for (int r = 0; r < 8; ++r) {
        int n = n0 + 8 * hi + r;
        float v = mrow[r] ? __uint_as_float(0xFF800000u) : acc[r];
        simBase[(size_t)n * QLEN + qc] = v;
        unsigned long long pk =
            ((unsigned long long)enc_f32(v) << 32) | (0xFFFFFFFFu - (unsigned)n);
        lbest = (pk > lbest) ? pk : lbest;
      }
      unsigned long long other = __shfl_xor(lbest, 16, 32);
      lbest = (other > lbest) ? other : lbest;
      if (hi == 0) atomicMax(&bestLds[qc], lbest);
    }
  }
  __syncthreads();
  for (int qq = tid; qq < QLEN; qq += 256)
    atomicMax(&best[(size_t)bh * QSLOT + qq], bestLds[qq]);
}

// ---------------- gather + elementwise multiply ----------------
__global__ void __launch_bounds__(256)
gather_kernel(const float* __restrict__ xhat, const float* __restrict__ keyb,
              const unsigned long long* __restrict__ best,
              float* __restrict__ gathered, float* __restrict__ tmp) {
  int gid = blockIdx.x * 256 + threadIdx.x;   // B*QLEN*CDIM threads
  int c = gid & 255;
  int bq = gid >> 8;
  int b = bq / QLEN;
  int qq = bq - b * QLEN;
  int h = c >> 5;
  unsigned long long pk = best[(size_t)(b * NHEAD + h) * QSLOT + qq];
  unsigned n = 0xFFFFFFFFu - (unsigned)(pk & 0xFFFFFFFFu);
  float g = xhat[((size_t)b * NMEM + n) * CDIM + c];
  gathered[gid] = g;
  tmp[gid] = g * keyb[gid];
}

// ---------------- h = tmp @ proj_w^T + proj_b ----------------
__global__ void __launch_bounds__(256)
proj_kernel(const float* __restrict__ tmp, const float* __restrict__ w,
            const float* __restrict__ bias, float* __restrict__ hbuf) {
  __shared__ float row[CDIM];
  size_t bq = blockIdx.x;
  int co = threadIdx.x;
  row[co] = tmp[bq * CDIM + co];
  __syncthreads();
  float s = 0.0f;
  const float* wr = w + (size_t)co * CDIM;
#pragma unroll 8
  for (int c = 0; c < CDIM; ++c) s = fmaf(row[c], wr[c], s);
  hbuf[bq * CDIM + co] = s + bias[co];
}

// ---------------- 1/max(||h||_{dim=Q}, 1e-12) per (b,c) ----------------
__global__ void __launch_bounds__(256)
colnorm_kernel(const float* __restrict__ hbuf, float* __restrict__ inv) {
  int gid = blockIdx.x * 256 + threadIdx.x;   // BATCH*CDIM
  if (gid >= BATCH * CDIM) return;
  int b = gid >> 8, c = gid & 255;
  float s = 0.0f;
  for (int qq = 0; qq < QLEN; ++qq) {
    float v = hbuf[((size_t)(b * QLEN + qq)) * CDIM + c];
    s = fmaf(v, v, s);
  }
  inv[gid] = 1.0f / fmaxf(sqrtf(s), 1e-12f);
}

// ---------------- out1 = h*inv*alpha + gathered ----------------
__global__ void __launch_bounds__(256)
out1_kernel(const float* __restrict__ hbuf, const float* __restrict__ inv,
            const float* __restrict__ alpha, const float* __restrict__ gathered,
            float* __restrict__ out1) {
  int gid = blockIdx.x * 256 + threadIdx.x;
  int c = gid & 255;
  int b = (gid >> 8) / QLEN;
  out1[gid] = hbuf[gid] * inv[b * CDIM + c] * alpha[c] + gathered[gid];
}

// ---------------- out = out1 @ final_w^T + final_b + qn ----------------
__global__ void __launch_bounds__(256)
final_kernel(const float* __restrict__ out1, const float* __restrict__ w,
             const float* __restrict__ bias, const float* __restrict__ qn,
             float* __restrict__ out) {
  __shared__ float row[CDIM];
  size_t bq = blockIdx.x;
  int co = threadIdx.x;
  row[co] = out1[bq * CDIM + co];
  __syncthreads();
  float s = 0.0f;
  const float* wr = w + (size_t)co * CDIM;
#pragma unroll 8
  for (int c = 0; c < CDIM; ++c) s = fmaf(row[c], wr[c], s);
  out[bq * CDIM + co] = s + bias[co] + qn[bq * CDIM + co];
}

extern "C" void kernel_launch(void* const* d_in, const int* in_sizes, int n_in,
                              void* d_out, int out_size, void* d_ws, size_t ws_size,
                              hipStream_t stream) {
  const float* q        = (const float*)d_in[0];
  const float* x        = (const float*)d_in[1];
  const float* qpos     = (const float*)d_in[2];
  const float* xpos     = (const float*)d_in[3];
  const unsigned char* attn_mask = (const unsigned char*)d_in[4];  // jax bool = 1 byte
  const float* gamma    = (const float*)d_in[5];
  const float* beta     = (const float*)d_in[6];
  const float* proj_w   = (const float*)d_in[7];
  const float* proj_b   = (const float*)d_in[8];
  const float* final_w  = (const float*)d_in[9];
  const float* final_b  = (const float*)d_in[10];
  const float* alpha    = (const float*)d_in[11];

  const size_t OUT1 = (size_t)BATCH * QLEN * CDIM;        // 204800 floats
  float* out = (float*)d_out;                             // [B,Q,C]
  float* sim = (float*)d_out + OUT1;                      // [B,H,N,Q]

  // workspace layout (bytes)
  const size_t XB = (size_t)BATCH * NMEM * CDIM * 4;      // 67 MB
  const size_t TB = OUT1 * 4;                             // 819200
  char* ws = (char*)d_ws;
  float* xhat      = (float*)(ws);
  float* qn        = (float*)(ws + XB);
  float* keyb      = (float*)(ws + XB + TB);
  float* gathered  = (float*)(ws + XB + 2 * TB);
  float* tmp       = (float*)(ws + XB + 3 * TB);
  float* hbuf      = (float*)(ws + XB + 4 * TB);
  float* out1      = (float*)(ws + XB + 5 * TB);
  float* inv       = (float*)(ws + XB + 6 * TB);
  unsigned long long* best =
      (unsigned long long*)(ws + XB + 6 * TB + (size_t)BATCH * CDIM * 4);

  prep_q_kernel<<<BATCH * QLEN, 256, 0, stream>>>(q, qpos, gamma, beta, qn, keyb);
  prep_x_kernel<<<BATCH * NMEM, 256, 0, stream>>>(x, xpos, xhat);
  hipMemsetAsync(best, 0, (size_t)BATCH * NHEAD * QSLOT * 8, stream);
  dim3 sgrid(BATCH * NHEAD, NMEM / ROWS_PER_WG);
  sim_kernel<<<sgrid, 256, 0, stream>>>(xhat, keyb, attn_mask, sim, best);
  gather_kernel<<<BATCH * QLEN, 256, 0, stream>>>(xhat, keyb, best, gathered, tmp);
  proj_kernel<<<BATCH * QLEN, 256, 0, stream>>>(tmp, proj_w, proj_b, hbuf);
  colnorm_kernel<<<(BATCH * CDIM + 255) / 256, 256, 0, stream>>>(hbuf, inv);
  out1_kernel<<<BATCH * QLEN, 256, 0, stream>>>(hbuf, inv, alpha, gathered, out1);
  final_kernel<<<BATCH * QLEN, 256, 0, stream>>>(out1, final_w, final_b, qn, out);
}